// WOSNetMnist_55576876810253
// MI455X (gfx1250) — compile-verified
//
#include <hip/hip_runtime.h>
#include <stdint.h>

#define TPB 256

// ---- weight offsets inside the LDS weight pool (floats) ----
#define OFF_C1 0
#define OFF_L1 288      // linc1, stored TRANSPOSED: [d][c]
#define OFF_C2 1312
#define OFF_L2 1600     // transposed
#define OFF_C3 2624
#define OFF_C4 2912
#define OFF_L3 3200     // transposed
#define OFF_L4 4224
#define W_TOTAL 4256

// ------------------------------------------------------------------
// CDNA5 async global->LDS DMA (gfx1250). the LDS address is a VGPR offset,
// so we pass the low 32 bits of the generic shared pointer.
// Inline asm keeps this portable across ROCm-7.2 / amdgpu-toolchain.
// ------------------------------------------------------------------
__device__ __forceinline__ void async_ld_f32(const float* lds_dst, const float* gsrc) {
  uint32_t loff = (uint32_t)(uintptr_t)lds_dst;
  asm volatile("global_load_async_to_lds_b32 %0, %1, off"
               :
               : "v"(loff), "v"(gsrc)
               : "memory");
}

__device__ __forceinline__ void wait_async0() {
#if __has_builtin(__builtin_amdgcn_s_wait_asynccnt)
  __builtin_amdgcn_s_wait_asynccnt(0);
#else
  asm volatile("s_wait_asynccnt 0" ::: "memory");
#endif
}

// ---- compare-exchange primitives ----
__device__ __forceinline__ void cswap(float& a, float& b) {
  float lo = __builtin_fminf(a, b);
  float hi = __builtin_fmaxf(a, b);
  a = lo; b = hi;
}

// Classic 19-comparator median-of-9 (Paeth).
__device__ __forceinline__ float median9(float p0, float p1, float p2,
                                         float p3, float p4, float p5,
                                         float p6, float p7, float p8) {
  cswap(p1,p2); cswap(p4,p5); cswap(p7,p8);
  cswap(p0,p1); cswap(p3,p4); cswap(p6,p7);
  cswap(p1,p2); cswap(p4,p5); cswap(p7,p8);
  cswap(p0,p3); cswap(p5,p8); cswap(p4,p7);
  cswap(p3,p6); cswap(p1,p4); cswap(p2,p5);
  cswap(p4,p7); cswap(p4,p2); cswap(p6,p4);
  cswap(p4,p2);
  return p4;
}

// Rank-16 (0-based) of 32 values: fully-unrolled bitonic sort, constant
// indices -> array lives in VGPRs. 240 comparators.
__device__ __forceinline__ float rank16_of_32(float z[32]) {
#pragma unroll
  for (int k = 2; k <= 32; k <<= 1) {
#pragma unroll
    for (int j = k >> 1; j > 0; j >>= 1) {
#pragma unroll
      for (int i = 0; i < 32; ++i) {
        int l = i ^ j;
        if (l > i) {
          if ((i & k) == 0) cswap(z[i], z[l]);
          else              cswap(z[l], z[i]);
        }
      }
    }
  }
  return z[16];
}

// In-place 32->32 WOS "linc" layer over NPIX pixels resident in buf[32][NPIX].
// Wave layout: lane = cout (channel reads broadcast), wave = pixel slot.
// Rounds partition pixels, so reads(round r+1) never touch writes(round r).
template <int NPIX>
__device__ __forceinline__ void linc32_inplace(float* buf, const float* wT, int tid) {
  const int c  = tid & 31;   // lane -> output channel
  const int pr = tid >> 5;   // wave -> pixel slot (0..7)
  const int rounds = (NPIX + 7) >> 3;
  for (int r = 0; r < rounds; ++r) {
    const int p = (r << 3) + pr;
    const bool act = p < NPIX;
    float res = 0.0f;
    if (act) {
      float z[32];
#pragma unroll
      for (int d = 0; d < 32; ++d)
        z[d] = buf[d * NPIX + p] + wT[d * 32 + c];   // wT conflict-free, buf broadcast
      res = rank16_of_32(z);
    }
    __syncthreads();               // all reads of this pixel block complete
    if (act) buf[c * NPIX + p] = res;
  }
  __syncthreads();
}

__global__ __launch_bounds__(TPB) void wosnet_fused(
    const float* __restrict__ x,
    const float* __restrict__ wc1, const float* __restrict__ wl1,
    const float* __restrict__ wc2, const float* __restrict__ wl2,
    const float* __restrict__ wc3, const float* __restrict__ wc4,
    const float* __restrict__ wl3, const float* __restrict__ wl4,
    float* __restrict__ out)
{
  __shared__ float sA[32 * 26 * 26];  // 21632 f32 (86.5 KB) main activation
  __shared__ float sB[32 * 12 * 12];  //  4608 f32
  __shared__ float sC[32 * 10 * 10];  //  3200 f32
  __shared__ float sD[32 * 4 * 4];    //   512 f32
  __shared__ float sIn[28 * 28];      //   784 f32
  __shared__ float sW[W_TOTAL];       //  4256 f32   (total LDS ~140 KB)

  const int tid = threadIdx.x;
  const int b   = blockIdx.x;

  // ================= stage 0: async DMA image + weights into LDS ============
  const float* xs = x + b * 784;
  for (int i = tid; i < 784; i += TPB) async_ld_f32(&sIn[i], xs + i);
  for (int i = tid; i < 288; i += TPB) async_ld_f32(&sW[OFF_C1 + i], wc1 + i);
  for (int i = tid; i < 288; i += TPB) async_ld_f32(&sW[OFF_C2 + i], wc2 + i);
  for (int i = tid; i < 288; i += TPB) async_ld_f32(&sW[OFF_C3 + i], wc3 + i);
  for (int i = tid; i < 288; i += TPB) async_ld_f32(&sW[OFF_C4 + i], wc4 + i);
  // transpose linc weights during the DMA: sW[off + d*32 + c] = wl[c*32 + d]
  for (int i = tid; i < 1024; i += TPB) { int c = i >> 5, d = i & 31; async_ld_f32(&sW[OFF_L1 + d * 32 + c], wl1 + i); }
  for (int i = tid; i < 1024; i += TPB) { int c = i >> 5, d = i & 31; async_ld_f32(&sW[OFF_L2 + d * 32 + c], wl2 + i); }
  for (int i = tid; i < 1024; i += TPB) { int c = i >> 5, d = i & 31; async_ld_f32(&sW[OFF_L3 + d * 32 + c], wl3 + i); }
  for (int i = tid; i < 32;   i += TPB) async_ld_f32(&sW[OFF_L4 + i], wl4 + i);
  wait_async0();
  __syncthreads();

  // ================= conv1: WOS 3x3, 1->32, 28x28 -> [32][26][26] ============
  for (int idx = tid; idx < 32 * 676; idx += TPB) {
    int c = idx / 676, p = idx - c * 676;
    int h = p / 26, w = p - h * 26;
    const float* wr = &sW[OFF_C1 + c * 9];
    const float* ip = &sIn[h * 28 + w];
    sA[idx] = median9(ip[0]  + wr[0], ip[1]  + wr[1], ip[2]  + wr[2],
                      ip[28] + wr[3], ip[29] + wr[4], ip[30] + wr[5],
                      ip[56] + wr[6], ip[57] + wr[7], ip[58] + wr[8]);
  }
  __syncthreads();

  // ================= linc1 (in place on sA, 676 pixels) ======================
  linc32_inplace<676>(sA, &sW[OFF_L1], tid);

  // ========== conv2 depthwise 3x3 + maxpool2: [32][26][26] -> [32][12][12] ===
  for (int idx = tid; idx < 32 * 144; idx += TPB) {
    int c = idx / 144, q = idx - c * 144;
    int ph = q / 12, pw = q - ph * 12;
    const float* wr   = &sW[OFF_C2 + c * 9];
    const float* base = &sA[c * 676];
    float m = -3.4e38f;
#pragma unroll
    for (int dy = 0; dy < 2; ++dy)
#pragma unroll
      for (int dx = 0; dx < 2; ++dx) {
        const float* ip = base + (2 * ph + dy) * 26 + (2 * pw + dx);
        float v = median9(ip[0]  + wr[0], ip[1]  + wr[1], ip[2]  + wr[2],
                          ip[26] + wr[3], ip[27] + wr[4], ip[28] + wr[5],
                          ip[52] + wr[6], ip[53] + wr[7], ip[54] + wr[8]);
        m = __builtin_fmaxf(m, v);
      }
    sB[idx] = m;
  }
  __syncthreads();

  // ================= linc2 (in place on sB, 144 pixels) ======================
  linc32_inplace<144>(sB, &sW[OFF_L2], tid);

  // ========== conv3 depthwise: [32][12][12] -> [32][10][10] ==================
  for (int idx = tid; idx < 32 * 100; idx += TPB) {
    int c = idx / 100, q = idx - c * 100;
    int h = q / 10, w = q - h * 10;
    const float* wr = &sW[OFF_C3 + c * 9];
    const float* ip = &sB[c * 144 + h * 12 + w];
    sC[idx] = median9(ip[0]  + wr[0], ip[1]  + wr[1], ip[2]  + wr[2],
                      ip[12] + wr[3], ip[13] + wr[4], ip[14] + wr[5],
                      ip[24] + wr[6], ip[25] + wr[7], ip[26] + wr[8]);
  }
  __syncthreads();

  // ====== conv4 depthwise + maxpool2: [32][10][10] -> 8x8 -> [32][4][4] ======
  for (int idx = tid; idx < 32 * 16; idx += TPB) {
    int c = idx >> 4, q = idx & 15;
    int ph = q >> 2, pw = q & 3;
    const float* wr   = &sW[OFF_C4 + c * 9];
    const float* base = &sC[c * 100];
    float m = -3.4e38f;
#pragma unroll
    for (int dy = 0; dy < 2; ++dy)
#pragma unroll
      for (int dx = 0; dx < 2; ++dx) {
        const float* ip = base + (2 * ph + dy) * 10 + (2 * pw + dx);
        float v = median9(ip[0]  + wr[0], ip[1]  + wr[1], ip[2]  + wr[2],
                          ip[10] + wr[3], ip[11] + wr[4], ip[12] + wr[5],
                          ip[20] + wr[6], ip[21] + wr[7], ip[22] + wr[8]);
        m = __builtin_fmaxf(m, v);
      }
    sD[idx] = m;
  }
  __syncthreads();

  // ================= linc3 (in place on sD, 16 pixels) =======================
  linc32_inplace<16>(sD, &sW[OFF_L3], tid);

  // ============ linc4 (32->1, 16 pixels) + final maxpool4 -> out[b] ==========
  {
    float res = 0.0f;
    const bool act = tid < 16;
    if (act) {
      float z[32];
#pragma unroll
      for (int d = 0; d < 32; ++d) z[d] = sD[d * 16 + tid] + sW[OFF_L4 + d];
      res = rank16_of_32(z);
    }
    if (act) sC[tid] = res;     // sC is dead storage at this point
    __syncthreads();
    if (tid == 0) {
      float m = sC[0];
#pragma unroll
      for (int i = 1; i < 16; ++i) m = __builtin_fmaxf(m, sC[i]);
      out[b] = m;
    }
  }
}

extern "C" void kernel_launch(void* const* d_in, const int* in_sizes, int n_in,
                              void* d_out, int out_size, void* d_ws, size_t ws_size,
                              hipStream_t stream) {
  (void)n_in; (void)d_ws; (void)ws_size; (void)out_size;
  const float* x   = (const float*)d_in[0];
  const float* wc1 = (const float*)d_in[1];
  const float* wl1 = (const float*)d_in[2];
  const float* wc2 = (const float*)d_in[3];
  const float* wl2 = (const float*)d_in[4];
  const float* wc3 = (const float*)d_in[5];
  const float* wc4 = (const float*)d_in[6];
  const float* wl3 = (const float*)d_in[7];
  const float* wl4 = (const float*)d_in[8];
  const int B = in_sizes[0] / 784;   // 256 images, one workgroup each
  wosnet_fused<<<dim3(B), dim3(TPB), 0, stream>>>(
      x, wc1, wl1, wc2, wl2, wc3, wc4, wl3, wl4, (float*)d_out);
}